// GLM_89524298318134
// MI455X (gfx1250) — compile-verified
//
#include <hip/hip_runtime.h>
#include <hip/hip_bf16.h>

typedef __attribute__((ext_vector_type(2))) float v2f;
typedef __attribute__((ext_vector_type(8))) float v8f;

// ---------------------------------------------------------------------------
// Kernel 1: causal FIR  conv(stim, k_stim)[0:T] * dt, shifted right by 1,
// computed with V_WMMA_F32_16X16X4_F32.
//   D[m][n] = sum_k stim[t0+m-k] * ks[k]   (all 16 columns identical)
// A 16x4 layout (ISA 7.12.2): lanes 0-15 -> M=lane, VGPR0=K0, VGPR1=K1;
//                             lanes16-31 -> M=lane-16, VGPR0=K2, VGPR1=K3.
// B 4x16 mirrors with N across lanes; taps are lane-group dependent only.
// C/D 16x16: column N==0 lives in lane 0 (M=0..7) and lane 16 (M=8..15).
// ---------------------------------------------------------------------------
__global__ void glm_stimconv_wmma(const float* __restrict__ stim,
                                  const float* __restrict__ kstim,
                                  const float* __restrict__ t,
                                  float* __restrict__ out,
                                  int T, int klen)
{
    __shared__ float ksPad[64];
    const int lane = threadIdx.x;            // one wave (32 threads) per block
    for (int i = lane; i < 64; i += 32)
        ksPad[i] = (i < klen) ? kstim[i] : 0.0f;
    __syncthreads();

    const float dt = t[1] - t[0];
    const int m  = lane & 15;
    const int kb = (lane >> 4) << 1;         // 0 for lanes 0-15, 2 for 16-31

    for (int it = 0; it < 16; ++it) {        // 16 output tiles per block
        const int t0 = (blockIdx.x * 16 + it) * 16;
        if (t0 >= T) break;
        v8f acc = {};
        #pragma unroll
        for (int c = 0; c < 16; ++c) {       // K = 64 taps in chunks of 4
            const int k0 = 4 * c + kb;
            const int i0 = t0 + m - k0;
            v2f a, b;
            a.x = (i0     >= 0 && i0     < T) ? stim[i0]     : 0.0f;
            a.y = (i0 - 1 >= 0 && i0 - 1 < T) ? stim[i0 - 1] : 0.0f;
            b.x = ksPad[k0];
            b.y = ksPad[k0 + 1];
            acc = __builtin_amdgcn_wmma_f32_16x16x4_f32(
                      false, a, false, b, (short)0, acc, false, false);
        }
        if (lane == 0 || lane == 16) {
            const int mb = (lane == 0) ? 0 : 8;
            #pragma unroll
            for (int g = 0; g < 8; ++g) {
                const int dst = t0 + mb + g + 1;   // shift_tensor(...,1,fill=0)
                if (dst < T) out[dst] = acc[g] * dt;
            }
        }
    }
    if (blockIdx.x == 0 && lane == 0) out[0] = 0.0f;
}

// ---------------------------------------------------------------------------
// Kernel 2: sequential GLM scan, one lane per trial.
// Spike test transformed to log-domain so that BOTH transcendentals depend
// only on the freshly loaded uniform r, leaving a 4-op carried chain:
//   add3(loglam) -> cmp -> cndmask -> fma
// ---------------------------------------------------------------------------
__global__ void glm_scan(const float* __restrict__ rnd,
                         const float* __restrict__ t,
                         const float* __restrict__ bias,
                         const float* __restrict__ hist_w,
                         const float* __restrict__ hist_tau,
                         const float* __restrict__ sconv,
                         float* __restrict__ loglam_out,
                         float* __restrict__ mask_out,
                         int T, int NTR)
{
    const int tr = blockIdx.x * blockDim.x + threadIdx.x;
    if (tr >= NTR) return;

    const float dt = t[1] - t[0];
    const float b  = bias[0];
    const float w0 = hist_w[0], w1 = hist_w[1];
    const float d0 = __expf(-dt / hist_tau[0]);
    const float d1 = __expf(-dt / hist_tau[1]);
    const float logdt = __logf(dt);

    float s0 = 0.0f, s1 = 0.0f;
    #pragma unroll 4
    for (int j = 0; j < T; ++j) {
        const size_t idx = (size_t)j * (size_t)NTR + (size_t)tr;
        const float sc = sconv[j];                     // uniform -> scalar load
        const float r  = rnd[idx];                     // coalesced row
        // p_spk > r  <=>  loglam > log(-log(1-r)) - log(dt)   (off the chain)
        const float lth = __logf(-__logf(1.0f - r)) - logdt;
        const float loglam = (b + sc) + s0 + s1;       // carried chain starts
        const bool  spk = loglam > lth;
        loglam_out[idx] = loglam;
        mask_out[idx]   = spk ? 1.0f : 0.0f;
        s0 = __builtin_fmaf(d0, s0, spk ? w0 : 0.0f);
        s1 = __builtin_fmaf(d1, s1, spk ? w1 : 0.0f);
    }
}

// ---------------------------------------------------------------------------
extern "C" void kernel_launch(void* const* d_in, const int* in_sizes, int n_in,
                              void* d_out, int out_size, void* d_ws, size_t ws_size,
                              hipStream_t stream)
{
    const float* t        = (const float*)d_in[0];
    const float* stim     = (const float*)d_in[1];
    const float* rnd      = (const float*)d_in[2];
    const float* bias     = (const float*)d_in[3];
    const float* k_stim   = (const float*)d_in[4];
    const float* hist_w   = (const float*)d_in[5];
    const float* hist_tau = (const float*)d_in[6];

    const int T    = in_sizes[0];            // 8192
    const int NTR  = in_sizes[2] / T;        // 1024
    const int KLEN = in_sizes[4];            // 50

    float* sconv  = (float*)d_ws;            // T floats of scratch
    float* loglam = (float*)d_out;           // (T, NTR)
    float* mask   = (float*)d_out + (size_t)T * (size_t)NTR;  // (T, NTR)

    // FIR via WMMA: 16 outputs per tile, 16 tiles per 1-wave block.
    const int numTiles  = (T + 15) / 16;
    const int convBlks  = (numTiles + 15) / 16;
    glm_stimconv_wmma<<<convBlks, 32, 0, stream>>>(stim, k_stim, t, sconv, T, KLEN);

    // Sequential scan: one lane per trial, 256-lane blocks for coalescing.
    const int blk = 256;
    const int grd = (NTR + blk - 1) / blk;
    glm_scan<<<grd, blk, 0, stream>>>(rnd, t, bias, hist_w, hist_tau,
                                      sconv, loglam, mask, T, NTR);
}